// TripletLoss_81827716923628
// MI455X (gfx1250) — compile-verified
//
#include <hip/hip_runtime.h>
#include <math.h>

// ---------------------------------------------------------------------------
// TripletLoss (hard mining) + CrossEntropy, fp32, MI455X / gfx1250.
//   d_in[0] embeddings f32 [4096,128]
//   d_in[1] labels     int [4096]
//   d_in[2] fc_w       f32 [100,128]
//   d_in[3] fc_b       f32 [100]
//   d_out   f32 [1]
// fp32 WMMA (V_WMMA_F32_16X16X4_F32) for both GEMMs; hard-mining and CE
// reductions fused into tile epilogues (no 64MB distance matrix); B tiles
// double-buffered in registers so loads retire under the WMMA chain.
// ---------------------------------------------------------------------------

typedef __attribute__((ext_vector_type(2))) float v2f;
typedef __attribute__((ext_vector_type(8))) float v8f;

#define BATCH    4096
#define DIM      128
#define NCLS     100
#define CTILES   7          // ceil(100/16) class tiles (padded to 112)
#define MARGIN_F 0.3f

__device__ __forceinline__ v8f wmma_f32(v2f a, v2f b, v8f c) {
  // 8 args: (neg_a, A, neg_b, B, c_mod, C, reuse_a, reuse_b)
  return __builtin_amdgcn_wmma_f32_16x16x4_f32(
      false, a, false, b, (short)0, c, false, false);
}

// Load one 16(x)x128(K) panel's worth of per-lane fragments: lane reads
// row (tileRow0 + n), elements kk*4 + hl*2 .. +1  (v2f, 8B, 16B stride).
__device__ __forceinline__ void load_tile(v2f (&b)[32], const float* base) {
#pragma unroll
  for (int kk = 0; kk < 32; ++kk)
    b[kk] = *(const v2f*)(base + kk * 4);
}

__device__ __forceinline__ v8f gemm_tile(const v2f (&a)[32], const v2f (&b)[32]) {
  v8f acc = {};
#pragma unroll
  for (int kk = 0; kk < 32; ++kk)
    acc = wmma_f32(a[kk], b[kk], acc);
  return acc;
}

// --------------------------- kernel 1: prep --------------------------------
__global__ void prep_kernel(const float* __restrict__ emb,
                            float* __restrict__ sq,
                            float* __restrict__ accum) {
  int i = blockIdx.x * blockDim.x + threadIdx.x;
  if (i == 0) { accum[0] = 0.0f; accum[1] = 0.0f; }
  if (i < BATCH) {
    const float4* p = (const float4*)(emb + (size_t)i * DIM);
    float s = 0.0f;
#pragma unroll
    for (int k = 0; k < DIM / 4; ++k) {
      float4 v = p[k];
      s += v.x * v.x + v.y * v.y + v.z * v.z + v.w * v.w;
    }
    sq[i] = s;
  }
}

// ----------------- kernel 2: Gram tiles + fused hard mining ----------------
__global__ void __launch_bounds__(128)
gram_triplet_kernel(const float* __restrict__ emb,
                    const int* __restrict__ labels,
                    const float* __restrict__ sq,
                    float* __restrict__ accum) {
  const int lane = threadIdx.x & 31;
  const int wave = threadIdx.x >> 5;
  const int hl   = lane >> 4;     // 16-lane half
  const int n    = lane & 15;
  const int i0   = (blockIdx.x * 4 + wave) * 16;

  // A fragments: lane holds row i0+n; VGPR0=K0+2*hl, VGPR1=K0+2*hl+1.
  v2f aF[32];
  load_tile(aF, emb + (size_t)(i0 + n) * DIM + hl * 2);

  float sqI[8];
  int   labI[8];
#pragma unroll
  for (int r = 0; r < 8; ++r) {
    int i = i0 + r + hl * 8;
    sqI[r]  = sq[i];
    labI[r] = labels[i];
  }

  float ap[8], an[8];
#pragma unroll
  for (int r = 0; r < 8; ++r) { ap[r] = -INFINITY; an[r] = INFINITY; }

  // Epilogue folded per tile: dist + hard-pos/neg running extrema.
  auto epilogue = [&](const v8f& acc, int j0) {
    const float sqJ  = sq[j0 + n];
    const int   labJ = labels[j0 + n];
#pragma unroll
    for (int r = 0; r < 8; ++r) {
      float d2   = sqI[r] + sqJ - 2.0f * acc[r];
      float dist = sqrtf(fmaxf(d2, 1e-12f));
      bool  same = (labI[r] == labJ);          // diag included, like reference
      ap[r] = same ? fmaxf(ap[r], dist) : ap[r];
      an[r] = same ? an[r] : fminf(an[r], dist);
    }
  };

  // Double-buffered B tiles: loads for tile jt+1 issue before the WMMA chain
  // of tile jt, so memory latency hides under matrix-pipe execution.
  v2f b0[32], b1[32];
  const float* bbase = emb + (size_t)n * DIM + hl * 2;   // tile jt: + jt*16*DIM
  load_tile(b0, bbase);
  for (int jt = 0; jt < BATCH / 16; jt += 2) {
    load_tile(b1, bbase + (size_t)(jt + 1) * 16 * DIM);
    v8f acc0 = gemm_tile(aF, b0);
    epilogue(acc0, jt * 16);

    int nj = (jt + 2 < BATCH / 16) ? (jt + 2) : 0;       // clamped prefetch
    load_tile(b0, bbase + (size_t)nj * 16 * DIM);
    v8f acc1 = gemm_tile(aF, b1);
    epilogue(acc1, (jt + 1) * 16);
  }

  // Butterfly reductions stay inside each 16-lane half (masks 1,2,4,8) —
  // exactly the per-row lane group of the f32 C layout.
  float lsum = 0.0f;
#pragma unroll
  for (int r = 0; r < 8; ++r) {
    float a = ap[r], b = an[r];
#pragma unroll
    for (int m = 1; m < 16; m <<= 1) {
      a = fmaxf(a, __shfl_xor(a, m, 32));
      b = fminf(b, __shfl_xor(b, m, 32));
    }
    lsum += fmaxf(a - b + MARGIN_F, 0.0f);
  }
  if (n == 0) atomicAdd(&accum[0], lsum);   // lanes 0 and 16: 8 rows each
}

// --------------- kernel 3: logits WMMA + fused cross-entropy ---------------
__global__ void __launch_bounds__(128)
logits_ce_kernel(const float* __restrict__ emb,
                 const int* __restrict__ labels,
                 const float* __restrict__ fc_w,
                 const float* __restrict__ fc_b,
                 float* __restrict__ accum) {
  const int lane = threadIdx.x & 31;
  const int wave = threadIdx.x >> 5;
  const int hl   = lane >> 4;
  const int n    = lane & 15;
  const int i0   = (blockIdx.x * 4 + wave) * 16;

  v2f aF[32];
  load_tile(aF, emb + (size_t)(i0 + n) * DIM + hl * 2);

  // Invalid class columns (c >= 100) just read row 0; their outputs are
  // masked to -inf in the CE epilogue, so no B zero-padding is needed.
  v8f acc[CTILES];
#pragma unroll
  for (int t = 0; t < CTILES; ++t) {
    const int c  = t * 16 + n;
    const int cc = (c < NCLS) ? c : 0;
    v2f bW[32];
    load_tile(bW, fc_w + (size_t)cc * DIM + hl * 2);
    acc[t] = gemm_tile(aF, bW);
  }

  float bias[CTILES];
#pragma unroll
  for (int t = 0; t < CTILES; ++t) {
    int c = t * 16 + n;
    bias[t] = (c < NCLS) ? fc_b[c] : 0.0f;
  }

  int labR[8];
#pragma unroll
  for (int r = 0; r < 8; ++r) labR[r] = labels[i0 + r + hl * 8];

  float csum = 0.0f;
#pragma unroll
  for (int r = 0; r < 8; ++r) {
    float x[CTILES];
    float m = -INFINITY;
#pragma unroll
    for (int t = 0; t < CTILES; ++t) {
      int c = t * 16 + n;
      x[t] = (c < NCLS) ? (acc[t][r] + bias[t]) : -INFINITY;
      m = fmaxf(m, x[t]);
    }
#pragma unroll
    for (int mm = 1; mm < 16; mm <<= 1)
      m = fmaxf(m, __shfl_xor(m, mm, 32));

    float s = 0.0f, g = 0.0f;
#pragma unroll
    for (int t = 0; t < CTILES; ++t) {
      int c = t * 16 + n;
      if (c < NCLS) {
        s += expf(x[t] - m);
        if (c == labR[r]) g += x[t];
      }
    }
#pragma unroll
    for (int mm = 1; mm < 16; mm <<= 1) {
      s += __shfl_xor(s, mm, 32);
      g += __shfl_xor(g, mm, 32);
    }
    csum += (m + logf(s)) - g;                  // lse - logit[label]
  }
  if (n == 0) atomicAdd(&accum[1], csum);
}

// --------------------------- kernel 4: finalize ----------------------------
__global__ void finalize_kernel(const float* __restrict__ accum,
                                float* __restrict__ out) {
  if (threadIdx.x == 0 && blockIdx.x == 0)
    out[0] = (accum[0] + accum[1]) * (1.0f / (float)BATCH);
}

// ---------------------------------------------------------------------------
extern "C" void kernel_launch(void* const* d_in, const int* in_sizes, int n_in,
                              void* d_out, int out_size, void* d_ws, size_t ws_size,
                              hipStream_t stream) {
  (void)in_sizes; (void)n_in; (void)out_size; (void)ws_size;
  const float* emb    = (const float*)d_in[0];
  const int*   labels = (const int*)d_in[1];
  const float* fc_w   = (const float*)d_in[2];
  const float* fc_b   = (const float*)d_in[3];
  float* out = (float*)d_out;

  float* sq    = (float*)d_ws;      // 4096 floats
  float* accum = sq + BATCH;        // [0]=triplet sum, [1]=ce sum

  prep_kernel<<<(BATCH + 255) / 256, 256, 0, stream>>>(emb, sq, accum);
  gram_triplet_kernel<<<BATCH / 64, 128, 0, stream>>>(emb, labels, sq, accum);
  logits_ce_kernel<<<BATCH / 64, 128, 0, stream>>>(emb, labels, fc_w, fc_b, accum);
  finalize_kernel<<<1, 32, 0, stream>>>(accum, out);
}